// MyOutGAT_687194767720
// MI455X (gfx1250) — compile-verified
//
#include <hip/hip_runtime.h>
#include <math.h>

typedef __attribute__((ext_vector_type(2))) float v2f;
typedef __attribute__((ext_vector_type(8))) float v8f;

#define EPS_BN 1e-5f
#define EPS_SM 1e-16f
#define NEG_ATT 0.2f
#define NEG_ACT 0.01f
#define NEGINF_ENC 0x007FFFFFu  // monotonic encoding of -inf

// ---------- helpers ----------
__device__ __forceinline__ unsigned f2mono(float x) {
  unsigned u = __float_as_uint(x);
  return (u & 0x80000000u) ? ~u : (u | 0x80000000u);
}
__device__ __forceinline__ float mono2f(unsigned u) {
  return (u & 0x80000000u) ? __uint_as_float(u ^ 0x80000000u) : __uint_as_float(~u);
}
__device__ __forceinline__ float lrelu(float v, float s) { return v > 0.f ? v : s * v; }

// ---------- fills ----------
__global__ void fill_f32_k(float* p, float v, size_t n) {
  size_t i = blockIdx.x * (size_t)blockDim.x + threadIdx.x;
  if (i < n) p[i] = v;
}
__global__ void fill_u32_k(unsigned* p, unsigned v, size_t n) {
  size_t i = blockIdx.x * (size_t)blockDim.x + threadIdx.x;
  if (i < n) p[i] = v;
}

// ---------- fp32 WMMA GEMM: C[M,Nc] = A[M,K] @ B[K,Nc], row-major ----------
// One wave computes one 16x16 tile; K stepped by 4 via V_WMMA_F32_16X16X4_F32.
__global__ void gemm_f32_wmma(const float* __restrict__ A, const float* __restrict__ B,
                              float* __restrict__ C, int M, int Nc, int K) {
  const int lane = threadIdx.x & 31;
  const int wave = (int)((blockIdx.x * (size_t)blockDim.x + threadIdx.x) >> 5);
  const int tilesN = Nc >> 4;
  const int tilesM = (M + 15) >> 4;
  const int tileM = wave / tilesN;
  const int tileN = wave - tileM * tilesN;
  if (tileM >= tilesM) return;  // wave-uniform exit

  const int row0 = tileM << 4;
  const int col0 = tileN << 4;
  const int mA = row0 + (lane & 15);
  const int hi = lane >> 4;               // 0: K pair {0,1}, 1: K pair {2,3}
  const bool mvalid = (mA < M);
  const int mAc = mvalid ? mA : 0;        // clamp; zero contribution below
  const int nB = col0 + (lane & 15);

  v8f acc = {};
  for (int k0 = 0; k0 < K; k0 += 4) {
    const int ka = k0 + hi * 2;
    v2f a = *(const v2f*)(A + (size_t)mAc * K + ka);
    if (!mvalid) { a.x = 0.f; a.y = 0.f; }
    v2f b;
    b.x = B[(size_t)ka * Nc + nB];
    b.y = B[(size_t)(ka + 1) * Nc + nB];
    acc = __builtin_amdgcn_wmma_f32_16x16x4_f32(false, a, false, b, (short)0, acc,
                                                false, false);
  }
  const int colC = col0 + (lane & 15);
  const int rbase = row0 + (hi ? 8 : 0);
#pragma unroll
  for (int r = 0; r < 8; ++r) {
    const int row = rbase + r;
    if (row < M) C[(size_t)row * Nc + colC] = acc[r];
  }
}

// ---------- per-node attention projections: a_src/a_dst [N,H] ----------
__global__ void att_proj(const float* __restrict__ xh, const float* __restrict__ att_src,
                         const float* __restrict__ att_dst, float* __restrict__ a_src,
                         float* __restrict__ a_dst, int Nn, int H, int C) {
  int n = blockIdx.x * blockDim.x + threadIdx.x;
  if (n >= Nn) return;
  for (int h = 0; h < H; ++h) {
    const float* row = xh + (size_t)n * H * C + (size_t)h * C;
    float ss = 0.f, sd = 0.f;
    for (int c = 0; c < C; ++c) {
      float v = row[c];
      ss += v * att_src[h * C + c];
      sd += v * att_dst[h * C + c];
    }
    a_src[(size_t)n * H + h] = ss;
    a_dst[(size_t)n * H + h] = sd;
  }
}

// ---------- segment max over dst (monotonic-uint atomicMax) ----------
__global__ void edge_max(const int* __restrict__ esrc, const int* __restrict__ edst, int E,
                         const float* __restrict__ a_src, const float* __restrict__ a_dst,
                         unsigned* __restrict__ Mx, int Etot, int H) {
  int e = blockIdx.x * blockDim.x + threadIdx.x;
  if (e >= Etot) return;
  int s, d;
  if (e < E) { s = esrc[e]; d = edst[e]; } else { s = d = e - E; }
  for (int h = 0; h < H; ++h) {
    float l = lrelu(a_src[(size_t)s * H + h] + a_dst[(size_t)d * H + h], NEG_ATT);
    atomicMax(&Mx[(size_t)d * H + h], f2mono(l));
  }
}

// ---------- segment sum of exp(logit - max) ----------
__global__ void edge_sum(const int* __restrict__ esrc, const int* __restrict__ edst, int E,
                         const float* __restrict__ a_src, const float* __restrict__ a_dst,
                         const unsigned* __restrict__ Mx, float* __restrict__ S, int Etot,
                         int H) {
  int e = blockIdx.x * blockDim.x + threadIdx.x;
  if (e >= Etot) return;
  int s, d;
  if (e < E) { s = esrc[e]; d = edst[e]; } else { s = d = e - E; }
  for (int h = 0; h < H; ++h) {
    float l = lrelu(a_src[(size_t)s * H + h] + a_dst[(size_t)d * H + h], NEG_ATT);
    float m = mono2f(Mx[(size_t)d * H + h]);
    atomicAdd(&S[(size_t)d * H + h], expf(l - m));
  }
}

// ---------- weighted scatter-add of messages: out[d] += alpha * xh[s] ----------
// grid = Etot blocks, blockDim = H*C (128 or 64), logC = log2(C)
__global__ void edge_agg(const int* __restrict__ esrc, const int* __restrict__ edst, int E,
                         const float* __restrict__ a_src, const float* __restrict__ a_dst,
                         const unsigned* __restrict__ Mx, const float* __restrict__ S,
                         const float* __restrict__ xh, float* __restrict__ out, int H,
                         int logC) {
  int e = blockIdx.x;
  int c = threadIdx.x;
  int HC = blockDim.x;
  int s, d;
  if (e < E) { s = esrc[e]; d = edst[e]; } else { s = d = e - E; }
  int h = c >> logC;
  float l = lrelu(a_src[(size_t)s * H + h] + a_dst[(size_t)d * H + h], NEG_ATT);
  float m = mono2f(Mx[(size_t)d * H + h]);
  float alpha = expf(l - m) / (S[(size_t)d * H + h] + EPS_SM);
  atomicAdd(&out[(size_t)d * HC + c], alpha * xh[(size_t)s * HC + c]);
}

// ---------- bias + leaky_relu in place + per-channel sum/sumsq ----------
// blockDim = HC; each block covers rowsPerBlock rows, 2 atomics per channel.
__global__ void post_act_stats(float* __restrict__ out, const float* __restrict__ bias,
                               float* __restrict__ sum, float* __restrict__ sumsq, int Nn,
                               int rowsPerBlock) {
  int c = threadIdx.x;
  int HC = blockDim.x;
  int r0 = blockIdx.x * rowsPerBlock;
  int rend = r0 + rowsPerBlock;
  if (rend > Nn) rend = Nn;
  float bc = bias[c];
  float s = 0.f, ss = 0.f;
  for (int r = r0; r < rend; ++r) {
    float v = out[(size_t)r * HC + c] + bc;
    v = lrelu(v, NEG_ACT);
    out[(size_t)r * HC + c] = v;
    s += v;
    ss += v * v;
  }
  atomicAdd(&sum[c], s);
  atomicAdd(&sumsq[c], ss);
}

// ---------- batch-norm apply in place ----------
__global__ void bn_apply(float* __restrict__ out, const float* __restrict__ sum,
                         const float* __restrict__ sumsq, const float* __restrict__ gamma,
                         const float* __restrict__ beta, int Nn, int cmask, size_t total) {
  size_t t = blockIdx.x * (size_t)blockDim.x + threadIdx.x;
  if (t >= total) return;
  int c = (int)(t & (size_t)cmask);
  float invN = 1.f / (float)Nn;
  float mu = sum[c] * invN;
  float var = sumsq[c] * invN - mu * mu;
  out[t] = (out[t] - mu) * rsqrtf(var + EPS_BN) * gamma[c] + beta[c];
}

// ---------- gather + MLP 64->32->16->5 + sigmoid ----------
__global__ void mlp_head(const float* __restrict__ h, const int* __restrict__ idx,
                         const float* __restrict__ mw1, const float* __restrict__ mb1,
                         const float* __restrict__ mw2, const float* __restrict__ mb2,
                         const float* __restrict__ mw3, const float* __restrict__ mb3,
                         float* __restrict__ out_logits, float* __restrict__ out_sig,
                         int nsel) {
  int i = blockIdx.x * blockDim.x + threadIdx.x;
  if (i >= nsel) return;
  const float* row = h + (size_t)idx[i] * 64;
  float r0[64];
#pragma unroll
  for (int k = 0; k < 64; ++k) r0[k] = row[k];
  float t1[32];
#pragma unroll
  for (int j = 0; j < 32; ++j) {
    float a = mb1[j];
#pragma unroll
    for (int k = 0; k < 64; ++k) a += r0[k] * mw1[k * 32 + j];
    t1[j] = lrelu(a, NEG_ACT);
  }
  float t2[16];
#pragma unroll
  for (int j = 0; j < 16; ++j) {
    float a = mb2[j];
#pragma unroll
    for (int k = 0; k < 32; ++k) a += t1[k] * mw2[k * 16 + j];
    t2[j] = lrelu(a, NEG_ACT);
  }
#pragma unroll
  for (int j = 0; j < 5; ++j) {
    float a = mb3[j];
#pragma unroll
    for (int k = 0; k < 16; ++k) a += t2[k] * mw3[k * 5 + j];
    out_logits[(size_t)i * 5 + j] = a;
    out_sig[(size_t)i * 5 + j] = 1.f / (1.f + expf(-a));
  }
}

// ---------- host side ----------
static inline int cdiv(long long a, long long b) { return (int)((a + b - 1) / b); }

extern "C" void kernel_launch(void* const* d_in, const int* in_sizes, int n_in, void* d_out,
                              int out_size, void* d_ws, size_t ws_size, hipStream_t stream) {
  (void)n_in; (void)out_size; (void)ws_size;
  const float* x   = (const float*)d_in[0];
  const int* eidx  = (const int*)d_in[1];
  const int* idx   = (const int*)d_in[2];
  const float* w0  = (const float*)d_in[3];
  const float* as0 = (const float*)d_in[4];
  const float* ad0 = (const float*)d_in[5];
  const float* b0  = (const float*)d_in[6];
  const float* g0  = (const float*)d_in[7];
  const float* be0 = (const float*)d_in[8];
  const float* w1  = (const float*)d_in[9];
  const float* as1 = (const float*)d_in[10];
  const float* ad1 = (const float*)d_in[11];
  const float* b1  = (const float*)d_in[12];
  const float* g1  = (const float*)d_in[13];
  const float* be1 = (const float*)d_in[14];
  const float* mw1 = (const float*)d_in[15];
  const float* mb1 = (const float*)d_in[16];
  const float* mw2 = (const float*)d_in[17];
  const float* mb2 = (const float*)d_in[18];
  const float* mw3 = (const float*)d_in[19];
  const float* mb3 = (const float*)d_in[20];

  const int Nn = in_sizes[0] / 128;
  const int E = in_sizes[1] / 2;
  const int nsel = in_sizes[2];
  const int Etot = E + Nn;
  const int* esrc = eidx;      // edge_index[0]
  const int* edst = eidx + E;  // edge_index[1]

  // workspace layout (floats)
  float* ws = (float*)d_ws;
  size_t off = 0;
  float* BUF_A = ws + off; off += (size_t)Nn * 128;  // XH0; later XH1 | OUT1
  float* OUT0  = ws + off; off += (size_t)Nn * 128;
  float* ASRC0 = ws + off; off += (size_t)Nn * 2;
  float* ADST0 = ws + off; off += (size_t)Nn * 2;
  unsigned* M0 = (unsigned*)(ws + off); off += (size_t)Nn * 2;
  float* S0    = ws + off; off += (size_t)Nn * 2;
  float* ASRC1 = ws + off; off += (size_t)Nn;
  float* ADST1 = ws + off; off += (size_t)Nn;
  unsigned* M1 = (unsigned*)(ws + off); off += (size_t)Nn;
  float* S1    = ws + off; off += (size_t)Nn;
  float* SUMS  = ws + off; off += 128;   // per-channel sum
  float* SUMSQ = ws + off; off += 128;   // per-channel sumsq
  float* XH0 = BUF_A;
  float* XH1 = BUF_A;                    // reuse after layer-0 aggregation
  float* OUT1 = BUF_A + (size_t)Nn * 64;

  float* out_logits = (float*)d_out;
  float* out_sig = out_logits + (size_t)nsel * 5;

#define FILLF(p, v, n) fill_f32_k<<<cdiv((long long)(n), 256), 256, 0, stream>>>((p), (v), (size_t)(n))
#define FILLU(p, v, n) fill_u32_k<<<cdiv((long long)(n), 256), 256, 0, stream>>>((p), (v), (size_t)(n))

  // ================= layer 0: GATConv(128 -> 64, heads=2) =================
  FILLF(OUT0, 0.f, (size_t)Nn * 128);
  FILLF(S0, 0.f, (size_t)Nn * 2);
  FILLF(SUMS, 0.f, 128); FILLF(SUMSQ, 0.f, 128);
  FILLU(M0, NEGINF_ENC, (size_t)Nn * 2);

  {  // XH0 = x @ w0  [Nn,128]
    long long waves = (long long)((Nn + 15) / 16) * (128 / 16);
    gemm_f32_wmma<<<cdiv(waves, 8), 256, 0, stream>>>(x, w0, XH0, Nn, 128, 128);
  }
  att_proj<<<cdiv(Nn, 256), 256, 0, stream>>>(XH0, as0, ad0, ASRC0, ADST0, Nn, 2, 64);
  edge_max<<<cdiv(Etot, 256), 256, 0, stream>>>(esrc, edst, E, ASRC0, ADST0, M0, Etot, 2);
  edge_sum<<<cdiv(Etot, 256), 256, 0, stream>>>(esrc, edst, E, ASRC0, ADST0, M0, S0, Etot, 2);
  edge_agg<<<Etot, 128, 0, stream>>>(esrc, edst, E, ASRC0, ADST0, M0, S0, XH0, OUT0, 2, 6);
  post_act_stats<<<cdiv(Nn, 512), 128, 0, stream>>>(OUT0, b0, SUMS, SUMSQ, Nn, 512);
  bn_apply<<<cdiv((long long)Nn * 128, 256), 256, 0, stream>>>(OUT0, SUMS, SUMSQ, g0, be0,
                                                               Nn, 127, (size_t)Nn * 128);

  // ================= layer 1: GATConv(128 -> 64, heads=1) =================
  FILLF(OUT1, 0.f, (size_t)Nn * 64);
  FILLF(S1, 0.f, (size_t)Nn);
  FILLF(SUMS, 0.f, 128); FILLF(SUMSQ, 0.f, 128);
  FILLU(M1, NEGINF_ENC, (size_t)Nn);

  {  // XH1 = OUT0 @ w1  [Nn,64]
    long long waves = (long long)((Nn + 15) / 16) * (64 / 16);
    gemm_f32_wmma<<<cdiv(waves, 8), 256, 0, stream>>>(OUT0, w1, XH1, Nn, 64, 128);
  }
  att_proj<<<cdiv(Nn, 256), 256, 0, stream>>>(XH1, as1, ad1, ASRC1, ADST1, Nn, 1, 64);
  edge_max<<<cdiv(Etot, 256), 256, 0, stream>>>(esrc, edst, E, ASRC1, ADST1, M1, Etot, 1);
  edge_sum<<<cdiv(Etot, 256), 256, 0, stream>>>(esrc, edst, E, ASRC1, ADST1, M1, S1, Etot, 1);
  edge_agg<<<Etot, 64, 0, stream>>>(esrc, edst, E, ASRC1, ADST1, M1, S1, XH1, OUT1, 1, 6);
  post_act_stats<<<cdiv(Nn, 512), 64, 0, stream>>>(OUT1, b1, SUMS, SUMSQ, Nn, 512);
  bn_apply<<<cdiv((long long)Nn * 64, 256), 256, 0, stream>>>(OUT1, SUMS, SUMSQ, g1, be1,
                                                              Nn, 63, (size_t)Nn * 64);

  // ================= MLP head =================
  mlp_head<<<cdiv(nsel, 256), 256, 0, stream>>>(OUT1, idx, mw1, mb1, mw2, mb2, mw3, mb3,
                                                out_logits, out_sig, nsel);
#undef FILLF
#undef FILLU
}